// ASTGCNBlock_15805479649938
// MI455X (gfx1250) — compile-verified
//
#include <hip/hip_runtime.h>
#include <math.h>

// ASTGCN block for MI455X (gfx1250), fp32 end-to-end.
// Heavy GEMMs use V_WMMA_F32_16X16X4_F32 (full fp32 matrix pipe).
// Dominant graph-conv GEMM: double-buffered async global->LDS staging of the
// A-operand (ASYNCcnt overlap), 2x2 register tiling per wave, conflict-free
// LDS layout (row stride 40 floats, 16B-aligned for b128 async writes).

typedef float v2f __attribute__((ext_vector_type(2)));
typedef float v8f __attribute__((ext_vector_type(8)));

#define B_ 16
#define N_ 1024
#define F_ 64
#define T_ 64
#define K_ 3
#define C_ 64
#define FT_ (F_*T_)   // 4096
#define KC_ 128       // K-chunk staged in LDS for the cheb GEMM
#define ASTR_ 40      // LDS row stride (floats): 160B = 16B-aligned, bank-safe

__device__ __forceinline__ float sigmoidf_(float v){ return 1.0f/(1.0f+expf(-v)); }

__device__ __forceinline__ v8f wmma4(v2f a, v2f b, v8f c){
  // (neg_a, A, neg_b, B, c_mod, C, reuse_a, reuse_b)
  return __builtin_amdgcn_wmma_f32_16x16x4_f32(false, a, false, b, (short)0, c, false, false);
}

// async copy 16B global -> LDS (per lane), tracked by ASYNCcnt
__device__ __forceinline__ void async_b128(unsigned int lds_off, const float* src){
  asm volatile("global_load_async_to_lds_b128 %0, %1, off"
               :: "v"(lds_off), "v"((unsigned long long)(size_t)src) : "memory");
}
__device__ __forceinline__ void wait_async_le4(){
  asm volatile("s_wait_asynccnt 4" ::: "memory");
}
__device__ __forceinline__ void wait_async0(){
  asm volatile("s_wait_asynccnt 0" ::: "memory");
}

// ---------------- temporal attention ----------------

// tmp1[b,t,f] = sum_n x[b,n,f,t] * U1[n]
__global__ void k_tmp1(const float* __restrict__ x, const float* __restrict__ U1,
                       float* __restrict__ tmp1){
  int t = threadIdx.x;                 // 64
  int f = blockIdx.x*4 + threadIdx.y;  // gridDim.x = 16, blockDim.y = 4
  int b = blockIdx.y;
  const float* xp = x + ((size_t)b*N_*F_ + f)*T_ + t;
  float acc = 0.f;
  for (int n=0;n<N_;++n) acc += xp[(size_t)n*FT_] * U1[n];
  tmp1[((size_t)b*T_ + t)*F_ + f] = acc;
}

// rhsT[b,n,t] = sum_f U3[f] * x[b,n,f,t]
__global__ void k_rhsT(const float* __restrict__ x, const float* __restrict__ U3,
                       float* __restrict__ rhs){
  int t = threadIdx.x; int n = blockIdx.x; int b = blockIdx.y;
  const float* xp = x + ((size_t)b*N_+n)*(size_t)FT_ + t;
  float acc = 0.f;
  for (int f=0; f<F_; ++f) acc += U3[f]*xp[(size_t)f*T_];
  rhs[((size_t)b*N_+n)*T_+t] = acc;
}

// lhsT[b,t,n] = sum_f tmp1[b,t,f] * U2[f,n]
__global__ void k_lhsT(const float* __restrict__ tmp1, const float* __restrict__ U2,
                       float* __restrict__ lhs){
  int n  = blockIdx.x*256 + threadIdx.x;  // gridDim.x = 4
  int bt = blockIdx.y;                    // b*T + t
  const float* tp = tmp1 + (size_t)bt*F_;
  float acc = 0.f;
  for (int f=0; f<F_; ++f) acc += tp[f]*U2[(size_t)f*N_+n];
  lhs[(size_t)bt*N_+n] = acc;
}

// prodT[b,t,s] = sum_n lhsT[b,t,n] * rhsT[b,n,s]
__global__ void k_prodT(const float* __restrict__ lhs, const float* __restrict__ rhs,
                        float* __restrict__ prod){
  int b = blockIdx.x;
  for (int idx = threadIdx.x; idx < T_*T_; idx += blockDim.x){
    int t = idx>>6, s = idx&63;
    const float* lp = lhs + ((size_t)b*T_+t)*N_;
    const float* rp = rhs + (size_t)b*N_*T_ + s;
    float acc = 0.f;
    for (int n=0;n<N_;++n) acc += lp[n]*rp[(size_t)n*T_];
    prod[((size_t)b*T_+t)*T_+s] = acc;
  }
}

// E = colsoftmax_i( sum_j V_e[k,j]*sig(prod[b,i,j]+b_e[i,j]) ); EW1[b,j]=sum_t E[b,j,t]W1[t]
__global__ void k_E(const float* __restrict__ prod, const float* __restrict__ b_e,
                    const float* __restrict__ V_e, const float* __restrict__ W1,
                    float* __restrict__ E, float* __restrict__ EW1){
  __shared__ float sp[T_*T_];
  __shared__ float er[T_*T_];
  int b = blockIdx.x;
  for (int idx=threadIdx.x; idx<T_*T_; idx+=blockDim.x)
    sp[idx] = sigmoidf_(prod[(size_t)b*T_*T_+idx] + b_e[idx]);
  __syncthreads();
  for (int idx=threadIdx.x; idx<T_*T_; idx+=blockDim.x){
    int i=idx>>6, kc=idx&63;
    float acc = 0.f;
    for (int j=0;j<T_;++j) acc += V_e[kc*T_+j]*sp[i*T_+j];
    er[i*T_+kc] = acc;
  }
  __syncthreads();
  if (threadIdx.x < T_){
    int kc = threadIdx.x;
    float m = -1e30f;
    for (int i=0;i<T_;++i) m = fmaxf(m, er[i*T_+kc]);
    float s = 0.f;
    for (int i=0;i<T_;++i) s += expf(er[i*T_+kc]-m);
    float inv = 1.0f/s;
    for (int i=0;i<T_;++i){
      float v = expf(er[i*T_+kc]-m)*inv;
      E[(size_t)b*T_*T_ + i*T_ + kc] = v;
      sp[i*T_+kc] = v;   // reuse sp to hold normalized E for EW1
    }
  }
  __syncthreads();
  if (threadIdx.x < T_){
    int j = threadIdx.x; float acc = 0.f;
    for (int t=0;t<T_;++t) acc += sp[j*T_+t]*W1[t];
    EW1[(size_t)b*T_+j] = acc;
  }
}

// ---------------- spatial attention (x_tat folded away) ----------------

// lhsF[b,n,f] = sum_j x[b,n,f,j]*EW1[b,j];  xW3[b,n,j] = sum_f W3[f]*x[b,n,f,j]
__global__ void k_xpass2(const float* __restrict__ x, const float* __restrict__ EW1,
                         const float* __restrict__ W3,
                         float* __restrict__ lhsF, float* __restrict__ xW3){
  __shared__ float xl[F_*(T_+1)];
  __shared__ float ew[T_];
  int n = blockIdx.x, b = blockIdx.y;
  int tid = threadIdx.x;  // 64
  const float* xp = x + ((size_t)b*N_+n)*(size_t)FT_;
  for (int f=0; f<F_; ++f) xl[f*(T_+1)+tid] = xp[(size_t)f*T_+tid];
  if (tid < T_) ew[tid] = EW1[(size_t)b*T_+tid];
  __syncthreads();
  { int f = tid; float acc=0.f;
    for (int j=0;j<T_;++j) acc += xl[f*(T_+1)+j]*ew[j];
    lhsF[((size_t)b*N_+n)*F_+f] = acc; }
  { int j = tid; float acc=0.f;
    for (int f=0; f<F_; ++f) acc += W3[f]*xl[f*(T_+1)+j];
    xW3[((size_t)b*N_+n)*T_+j] = acc; }
}

// lhs2[b,n,t] = sum_f lhsF[b,n,f]*W2[f,t]
__global__ void k_lhs2(const float* __restrict__ lhsF, const float* __restrict__ W2,
                       float* __restrict__ lhs2){
  int t = threadIdx.x & 63;
  int n = blockIdx.x*4 + (threadIdx.x>>6);
  int b = blockIdx.y;
  const float* lp = lhsF + ((size_t)b*N_+n)*F_;
  float acc=0.f;
  for (int f=0; f<F_; ++f) acc += lp[f]*W2[f*T_+t];
  lhs2[((size_t)b*N_+n)*T_+t] = acc;
}

// rhs2[b,t,n] = sum_j xW3[b,n,j]*E[b,j,t]
__global__ void k_rhs2(const float* __restrict__ xW3, const float* __restrict__ E,
                       float* __restrict__ rhs2){
  int t = threadIdx.x & 63;
  int n = blockIdx.x*4 + (threadIdx.x>>6);
  int b = blockIdx.y;
  const float* xp = xW3 + ((size_t)b*N_+n)*T_;
  float acc=0.f;
  for (int j=0;j<T_;++j) acc += xp[j]*E[(size_t)b*T_*T_ + j*T_ + t];
  rhs2[(size_t)b*T_*N_ + (size_t)t*N_ + n] = acc;
}

// prod2[b,i,j] = sum_t lhs2[b,i,t]*rhs2[b,t,j]   (WMMA f32, one 16x16 tile per wave)
__global__ void k_prod2(const float* __restrict__ lhs2, const float* __restrict__ rhs2,
                        float* __restrict__ prod2){
  int wave = threadIdx.x >> 5, lane = threadIdx.x & 31;
  int L = lane & 15, h = lane >> 4;
  int tile = blockIdx.x*8 + wave;      // 4096 tiles per batch -> gridDim.x = 512
  int b = blockIdx.y;
  int i0 = (tile >> 6) * 16;
  int j0 = (tile & 63) * 16;
  const float* A  = lhs2 + (size_t)b*N_*T_;
  const float* Bm = rhs2 + (size_t)b*T_*N_;
  v8f acc = {};
  for (int t0 = 0; t0 < T_; t0 += 4){
    int tA = t0 + 2*h;
    v2f a, bb;
    a.x  = A[(size_t)(i0+L)*T_ + tA];
    a.y  = A[(size_t)(i0+L)*T_ + tA+1];
    bb.x = Bm[(size_t)tA*N_     + j0+L];
    bb.y = Bm[(size_t)(tA+1)*N_ + j0+L];
    acc = wmma4(a, bb, acc);
  }
  float* O = prod2 + (size_t)b*N_*N_;
  #pragma unroll
  for (int g=0; g<8; ++g) O[(size_t)(i0+g+8*h)*N_ + j0+L] = acc[g];
}

// Sraw[b,i,k] = sum_j sig(prod2[b,i,j]+b_s[i,j]) * V_s[k,j]
// WMMA f32; one A fragment (2 sigmoids) feeds 4 kc-tiles (register blocking).
__global__ void k_Sraw(const float* __restrict__ prod2, const float* __restrict__ b_s,
                       const float* __restrict__ V_s, float* __restrict__ Sraw){
  int wave = threadIdx.x >> 5, lane = threadIdx.x & 31;
  int L = lane & 15, h = lane >> 4;
  int tile = blockIdx.x*8 + wave;      // 1024 wave-jobs per batch -> gridDim.x = 128
  int b = blockIdx.y;
  int i0 = (tile >> 4) * 16;           // 64 i-tiles
  int k0 = (tile & 15) * 64;           // 16 kc-quads (4 tiles each)
  const float* P = prod2 + (size_t)b*N_*N_;
  v8f a0={}, a1={}, a2={}, a3={};
  for (int j0 = 0; j0 < N_; j0 += 4){
    int jA = j0 + 2*h;
    size_t ra = (size_t)(i0+L)*N_ + jA;
    v2f a;
    a.x = sigmoidf_(P[ra]   + b_s[ra]);
    a.y = sigmoidf_(P[ra+1] + b_s[ra+1]);
    const float* vp = V_s + (size_t)(k0+L)*N_ + jA;
    v2f q0,q1,q2,q3;
    q0.x = vp[0];               q0.y = vp[1];
    q1.x = vp[16*(size_t)N_];   q1.y = vp[16*(size_t)N_+1];
    q2.x = vp[32*(size_t)N_];   q2.y = vp[32*(size_t)N_+1];
    q3.x = vp[48*(size_t)N_];   q3.y = vp[48*(size_t)N_+1];
    a0 = wmma4(a,q0,a0); a1 = wmma4(a,q1,a1);
    a2 = wmma4(a,q2,a2); a3 = wmma4(a,q3,a3);
  }
  float* O = Sraw + (size_t)b*N_*N_;
  #pragma unroll
  for (int g=0; g<8; ++g){
    size_t row = (size_t)(i0+g+8*h)*N_;
    O[row + k0      + L] = a0[g];
    O[row + k0 + 16 + L] = a1[g];
    O[row + k0 + 32 + L] = a2[g];
    O[row + k0 + 48 + L] = a3[g];
  }
}

// column softmax over axis i (dim 1) of Sraw -> S
__global__ void k_softS(const float* __restrict__ Sraw, float* __restrict__ S){
  int kc = blockIdx.x*blockDim.x + threadIdx.x;  // gridDim.x = 4, block 256
  int b = blockIdx.y;
  const float* P = Sraw + (size_t)b*N_*N_ + kc;
  float m = -1e30f;
  for (int i=0;i<N_;++i) m = fmaxf(m, P[(size_t)i*N_]);
  float s = 0.f;
  for (int i=0;i<N_;++i) s += expf(P[(size_t)i*N_]-m);
  float inv = 1.0f/s;
  float* O = S + (size_t)b*N_*N_ + kc;
  for (int i=0;i<N_;++i) O[(size_t)i*N_] = expf(P[(size_t)i*N_]-m)*inv;
}

// ---------------- Chebyshev graph conv (dominant GEMM) ----------------

// Am[b,m,n] = cheb_k[m,n] * S[b,m,n]   (bandwidth pass; removes VALU mul + one
// load stream from the hot GEMM loop)
__global__ void k_amask(const float* __restrict__ cheb, const float* __restrict__ S,
                        float* __restrict__ Am, int kk){
  int b = blockIdx.y;
  size_t e = ((size_t)blockIdx.x*blockDim.x + threadIdx.x)*4;
  const float4 c4 = *(const float4*)(cheb + (size_t)kk*N_*N_ + e);
  const float4 s4 = *(const float4*)(S    + (size_t)b*N_*N_  + e);
  float4 o; o.x=c4.x*s4.x; o.y=c4.y*s4.y; o.z=c4.z*s4.z; o.w=c4.w*s4.w;
  *(float4*)(Am + (size_t)b*N_*N_ + e) = o;
}

// r[b,n,ft] = sum_m Am[b,m,n] * x[b,m,ft]
// Block covers 32 n rows x 256 ft cols; A slab (KC_ x 32) is double-buffered in
// LDS via async global->LDS (fill of chunk c+1 overlaps WMMA on chunk c).
// Each wave: 2x2 tile grid -> each A/B fragment feeds two WMMAs.
__global__ void k_cheb(const float* __restrict__ Am, const float* __restrict__ x,
                       float* __restrict__ r){
  __shared__ float Alds[2][KC_*ASTR_];   // 2 x 20KB
  int wave = threadIdx.x >> 5, lane = threadIdx.x & 31;
  int L = lane & 15, h = lane >> 4;
  int b   = blockIdx.y;
  int n0  = (blockIdx.x >> 4) * 32;            // 32 n-blocks
  int ftb = (blockIdx.x & 15) * 256 + wave*32; // 16 ft-blocks; wave covers 32 cols
  const float* Ab = Am + (size_t)b*N_*N_;
  const float* Xb = x  + (size_t)b*N_*FT_;

  auto fill = [&](int bufi, int c){
    #pragma unroll
    for (int i = 0; i < 4; ++i){
      int e = ((int)threadIdx.x + i*256)*4;    // data float idx in slab [0,4096)
      int mrow = e >> 5, col = e & 31;         // 32 data cols per row
      const float* src = Ab + (size_t)(c + mrow)*N_ + n0 + col;
      unsigned int dst = (unsigned int)(size_t)&Alds[bufi][mrow*ASTR_ + col];
      async_b128(dst, src);                    // 16B-aligned (ASTR_*4 = 160B rows)
    }
  };

  v8f a00={}, a01={}, a10={}, a11={};
  fill(0, 0);
  for (int c = 0; c < N_; c += KC_){
    int buf = (c/KC_) & 1;
    bool more = (c + KC_ < N_);
    if (more) fill(buf^1, c + KC_);
    if (more) wait_async_le4(); else wait_async0();  // current slab resident
    __syncthreads();
    const float* Al = Alds[buf];
    for (int m0 = 0; m0 < KC_; m0 += 4){
      int mL = m0 + 2*h;
      v2f a0, a1;
      a0.x = Al[mL*ASTR_ + L];          a0.y = Al[(mL+1)*ASTR_ + L];
      a1.x = Al[mL*ASTR_ + 16 + L];     a1.y = Al[(mL+1)*ASTR_ + 16 + L];
      const float* xr = Xb + (size_t)(c + mL)*FT_ + ftb + L;
      v2f b0, b1;
      b0.x = xr[0];  b0.y = xr[FT_];
      b1.x = xr[16]; b1.y = xr[16+FT_];
      a00 = wmma4(a0,b0,a00); a01 = wmma4(a0,b1,a01);
      a10 = wmma4(a1,b0,a10); a11 = wmma4(a1,b1,a11);
    }
    __syncthreads();   // all waves done with buf before it is refilled
  }
  float* O = r + (size_t)b*N_*FT_;
  #pragma unroll
  for (int g=0; g<8; ++g){
    size_t r0 = (size_t)(n0      + g + 8*h)*FT_ + ftb + L;
    size_t r1 = (size_t)(n0 + 16 + g + 8*h)*FT_ + ftb + L;
    O[r0]      = a00[g];
    O[r0 + 16] = a01[g];
    O[r1]      = a10[g];
    O[r1 + 16] = a11[g];
  }
}

// out_acc[b,n,o,t] (+)= sum_f r[b,n,f,t]*Theta[k][f,o]   (WMMA f32, accumulate via C)
__global__ void k_theta(const float* __restrict__ r, const float* __restrict__ Theta,
                        float* __restrict__ out_acc, int kk, int accum){
  int wave = threadIdx.x >> 5, lane = threadIdx.x & 31;
  int L = lane & 15, h = lane >> 4;
  int bn = blockIdx.x;                 // B*N blocks
  const float* Rb = r + (size_t)bn*FT_;
  const float* Th = Theta + (size_t)kk*F_*C_;
  float* Ob = out_acc + (size_t)bn*C_*T_;
  for (int rep=0; rep<2; ++rep){
    int tid = wave + rep*8;            // 16 tiles (4 o-tiles x 4 t-tiles)
    int o0 = (tid >> 2)*16, t0 = (tid & 3)*16;
    v8f acc = {};
    if (accum){
      #pragma unroll
      for (int g=0; g<8; ++g) acc[g] = Ob[(size_t)(o0+g+8*h)*T_ + t0+L];
    }
    for (int f0=0; f0<F_; f0+=4){
      int fA = f0 + 2*h;
      v2f a, bb;
      a.x  = Th[(size_t)fA*C_     + o0+L];
      a.y  = Th[(size_t)(fA+1)*C_ + o0+L];
      bb.x = Rb[(size_t)fA*T_     + t0+L];
      bb.y = Rb[(size_t)(fA+1)*T_ + t0+L];
      acc = wmma4(a, bb, acc);
    }
    #pragma unroll
    for (int g=0; g<8; ++g) Ob[(size_t)(o0+g+8*h)*T_ + t0+L] = acc[g];
  }
}

// ---------------- fused tail: relu -> (1,3) conv -> 1x1 residual -> relu -> LN ----------------

__global__ void k_final(const float* __restrict__ x, const float* __restrict__ out_acc,
                        const float* __restrict__ tw, const float* __restrict__ tb,
                        const float* __restrict__ rw, const float* __restrict__ rb,
                        const float* __restrict__ gamma, const float* __restrict__ beta,
                        float* __restrict__ out){
  __shared__ float xl[F_*(T_+1)];
  __shared__ float sg[C_*(T_+1)];
  __shared__ float yl[C_*(T_+1)];
  __shared__ float mu[C_], rs[C_];
  int n = blockIdx.x, b = blockIdx.y;
  size_t base = ((size_t)b*N_+n)*(size_t)FT_;   // C*T == F*T
  for (int idx=threadIdx.x; idx<FT_; idx+=blockDim.x){
    int f = idx>>6, t = idx&63;
    xl[f*(T_+1)+t] = x[base+idx];
    float v = out_acc[base+idx];
    sg[f*(T_+1)+t] = v>0.f ? v : 0.f;
  }
  __syncthreads();
  int t  = threadIdx.x & 63;
  int og = threadIdx.x >> 6;   // 0..3
  for (int rep=0; rep<16; ++rep){
    int o = rep*4 + og;
    float tc = tb[o];
    for (int i=0;i<C_;++i){
      const float* w = tw + ((size_t)o*C_ + i)*3;
      float s1 = sg[i*(T_+1)+t];
      float s0 = (t>0)  ? sg[i*(T_+1)+t-1] : 0.f;
      float s2 = (t<63) ? sg[i*(T_+1)+t+1] : 0.f;
      tc += w[0]*s0 + w[1]*s1 + w[2]*s2;
    }
    float res = rb[o];
    for (int f=0; f<F_; ++f) res += rw[(size_t)o*F_+f]*xl[f*(T_+1)+t];
    float y = res + tc;
    yl[o*(T_+1)+t] = y>0.f ? y : 0.f;
  }
  __syncthreads();
  if (threadIdx.x < C_){
    int o = threadIdx.x;
    float m=0.f;
    for (int tt=0;tt<T_;++tt) m += yl[o*(T_+1)+tt];
    m *= (1.0f/T_);
    float v=0.f;
    for (int tt=0;tt<T_;++tt){ float d = yl[o*(T_+1)+tt]-m; v += d*d; }
    v *= (1.0f/T_);
    mu[o]=m; rs[o]=rsqrtf(v+1e-5f);
  }
  __syncthreads();
  for (int idx=threadIdx.x; idx<C_*T_; idx+=blockDim.x){
    int o = idx>>6, tt = idx&63;
    // gamma/beta (size C==T) broadcast over LAST axis (t), per reference semantics
    out[base+idx] = (yl[o*(T_+1)+tt]-mu[o])*rs[o]*gamma[tt] + beta[tt];
  }
}

extern "C" void kernel_launch(void* const* d_in, const int* in_sizes, int n_in,
                              void* d_out, int out_size, void* d_ws, size_t ws_size,
                              hipStream_t stream) {
  (void)in_sizes; (void)n_in; (void)out_size; (void)ws_size;
  const float* x    = (const float*)d_in[0];
  const float* cheb = (const float*)d_in[1];
  const float* Theta= (const float*)d_in[2];
  const float* W1   = (const float*)d_in[3];
  const float* W2   = (const float*)d_in[4];
  const float* W3   = (const float*)d_in[5];
  const float* b_s  = (const float*)d_in[6];
  const float* V_s  = (const float*)d_in[7];
  const float* U1   = (const float*)d_in[8];
  const float* U2   = (const float*)d_in[9];
  const float* U3   = (const float*)d_in[10];
  const float* b_e  = (const float*)d_in[11];
  const float* V_e  = (const float*)d_in[12];
  const float* tw   = (const float*)d_in[13];
  const float* tb   = (const float*)d_in[14];
  const float* rw   = (const float*)d_in[15];
  const float* rb   = (const float*)d_in[16];
  const float* gamma= (const float*)d_in[17];
  const float* beta = (const float*)d_in[18];
  float* out = (float*)d_out;

  float* w = (float*)d_ws;
  size_t off = 0;
  float* tmp1 = w + off;  off += (size_t)B_*T_*F_;
  float* rhsT = w + off;  off += (size_t)B_*N_*T_;
  float* lhsT = w + off;  off += (size_t)B_*T_*N_;
  float* prodT= w + off;  off += (size_t)B_*T_*T_;
  float* E    = w + off;  off += (size_t)B_*T_*T_;
  float* EW1  = w + off;  off += (size_t)B_*T_;
  float* lhsF = w + off;  off += (size_t)B_*N_*F_;
  float* xW3  = w + off;  off += (size_t)B_*N_*T_;
  float* lhs2 = w + off;  off += (size_t)B_*N_*T_;
  float* rhs2 = w + off;  off += (size_t)B_*T_*N_;
  float* prod2= w + off;  off += (size_t)B_*N_*N_;
  float* Sraw = w + off;  off += (size_t)B_*N_*N_;
  float* Sm   = prod2;    // alias: prod2 dead after Sraw; softmax writes S here
  float* Amask= Sraw;     // alias: Sraw dead after softmax; masked adjacency per k
  float* rbuf = w + off;  off += (size_t)B_*N_*FT_;
  float* oacc = w + off;  off += (size_t)B_*N_*C_*T_;

  // temporal attention
  k_tmp1 <<<dim3(16,B_),   dim3(64,4), 0, stream>>>(x, U1, tmp1);
  k_rhsT <<<dim3(N_,B_),   64,  0, stream>>>(x, U3, rhsT);
  k_lhsT <<<dim3(4,B_*T_), 256, 0, stream>>>(tmp1, U2, lhsT);
  k_prodT<<<B_,            256, 0, stream>>>(lhsT, rhsT, prodT);
  k_E    <<<B_,            256, 0, stream>>>(prodT, b_e, V_e, W1, E, EW1);
  // spatial attention (x_tat never materialized)
  k_xpass2<<<dim3(N_,B_),   64,  0, stream>>>(x, EW1, W3, lhsF, xW3);
  k_lhs2 <<<dim3(N_/4,B_), 256, 0, stream>>>(lhsF, W2, lhs2);
  k_rhs2 <<<dim3(N_/4,B_), 256, 0, stream>>>(xW3, E, rhs2);
  k_prod2<<<dim3(512,B_),  256, 0, stream>>>(lhs2, rhs2, prod2);
  k_Sraw <<<dim3(128,B_),  256, 0, stream>>>(prod2, b_s, V_s, Sraw);
  k_softS<<<dim3(4,B_),    256, 0, stream>>>(Sraw, Sm);
  // Chebyshev graph conv + Theta (dominant compute, fp32 WMMA + async double-buffered LDS)
  for (int kk=0; kk<K_; ++kk){
    k_amask<<<dim3(N_*N_/1024,B_), 256, 0, stream>>>(cheb, Sm, Amask, kk);
    k_cheb <<<dim3(512,B_),        256, 0, stream>>>(Amask, x, rbuf);
    k_theta<<<B_*N_,               256, 0, stream>>>(rbuf, Theta, oacc, kk, kk==0 ? 0 : 1);
  }
  // fused tail
  k_final<<<dim3(N_,B_), 256, 0, stream>>>(x, oacc, tw, tb, rw, rb, gamma, beta, out);
}